// LFS_Head_20315195310837
// MI455X (gfx1250) — compile-verified
//
#include <hip/hip_runtime.h>
#include <math.h>

// MI455X / gfx1250, wave32. Fused LFS-head kernel:
//   per patch-row block: build P[100][128] in LDS (gray inline),
//   dct = G(100x100) @ P via chained V_WMMA_F32_16X16X4_F32 (f32 precision),
//   logmag + 6-band filter reduction in VALU, shuffle-combine, store.

typedef float v2f __attribute__((ext_vector_type(2)));
typedef float v8f __attribute__((ext_vector_type(8)));

#define S_WIN   10
#define MBANDS  6
#define IMG     256
#define NIMG    16
#define HP      128          // patch grid 128x128 (stride 2, pad 4, win 10)
#define KLPAD   112          // 100 freq rows padded to 7*16
#define GPITCH  101          // odd pitch -> conflict-free strided row access
#define PPITCH  132          // 128 + pad
#define NTHREADS 256

__global__ __launch_bounds__(NTHREADS)
void lfs_head_kernel(const float* __restrict__ x,
                     const float* __restrict__ filt_learn,
                     float* __restrict__ out)
{
    extern __shared__ float smem[];
    float* Dm = smem;                       // [10][10] DCT matrix
    float* G  = Dm + 100;                   // [112][101] freq-gen matrix (padded)
    float* F  = G + KLPAD * GPITCH;         // [6][112] band filter / ft_num (padded w/ 0)
    float* P  = F + MBANDS * KLPAD;         // [100][132] patch pixels x patch-col

    const int tid = threadIdx.x;
    const int n   = blockIdx.x >> 7;        // image
    const int h   = blockIdx.x & 127;       // patch row

    // ---- DCT matrix D[k][m] ----
    if (tid < 100) {
        int k = tid / 10, m = tid % 10;
        float v = (k == 0)
            ? sqrtf(1.0f / S_WIN)
            : sqrtf(2.0f / S_WIN) *
              cosf(3.14159265358979323846f * (float)((2 * m + 1) * k) / (2.0f * S_WIN));
        Dm[tid] = v;
    }
    __syncthreads();

    // ---- G[(k,l)][(a,b)] = D[k][a]*D[l][b], rows 100..111 zero ----
    for (int idx = tid; idx < KLPAD * 100; idx += NTHREADS) {
        int kl = idx / 100, ab = idx % 100;
        float g = 0.0f;
        if (kl < 100)
            g = Dm[(kl / 10) * 10 + (ab / 10)] * Dm[(kl % 10) * 10 + (ab % 10)];
        G[kl * GPITCH + ab] = g;
    }

    // ---- F[m][kl] = (mask + sigma(learn)) / ft_num, padded cols zero ----
    for (int idx = tid; idx < MBANDS * KLPAD; idx += NTHREADS) {
        int m = idx / KLPAD, kl = idx % KLPAD;
        float f = 0.0f;
        if (kl < 100) {
            int k = kl / 10, l = kl % 10;
            int st = (2 * S_WIN * m) / MBANDS;       // matches Python int() floor
            int en = (2 * S_WIN * (m + 1)) / MBANDS;
            int freq = k + l;
            float mask = (freq >= st && freq < en) ? 1.0f : 0.0f;
            int cnt = 0;
            for (int i = 0; i < S_WIN; ++i)
                for (int j = 0; j < S_WIN; ++j)
                    if (i + j >= st && i + j < en) ++cnt;
            float e   = expf(-filt_learn[(m * S_WIN + k) * S_WIN + l]);
            float sig = (1.0f - e) / (1.0f + e);
            f = (mask + sig) / (float)cnt;
        }
        F[idx] = f;
    }

    // ---- P[ab][w]: patch pixels, gray computed inline, zero-padded borders ----
    const float* xn = x + (size_t)n * IMG * IMG * 3;
    for (int idx = tid; idx < 100 * HP; idx += NTHREADS) {
        int ab = idx / HP, w = idx % HP;
        int a = ab / 10, b = ab % 10;
        int yy = 2 * h - 4 + a;
        int xx = 2 * w - 4 + b;
        float g = 0.0f;
        if ((unsigned)yy < IMG && (unsigned)xx < IMG) {
            const float* px = xn + ((size_t)yy * IMG + xx) * 3;
            g = 127.5f * (0.299f * px[0] + 0.587f * px[1] + 0.114f * px[2]) + 127.5f;
        }
        P[ab * PPITCH + w] = g;
    }
    __syncthreads();

    // ---- WMMA: 7 freq-tiles x (K=100 as 25 steps of 4) per wave's 16 patches ----
    const int wave = tid >> 5;
    const int lane = tid & 31;
    const int col  = lane & 15;     // N / M position within tile
    const int hh   = lane >> 4;     // lane half
    const int wloc = wave * 16 + col;

    float sum[MBANDS] = {0.f, 0.f, 0.f, 0.f, 0.f, 0.f};

    for (int mt = 0; mt < 7; ++mt) {
        v8f acc = {0.f, 0.f, 0.f, 0.f, 0.f, 0.f, 0.f, 0.f};
        const float* Grow = &G[(mt * 16 + col) * GPITCH];
        #pragma unroll
        for (int kk = 0; kk < 25; ++kk) {
            // A(16x4 f32): lane half splits K: comp v -> K = 4*kk + 2*hh + v
            // B(4x16 f32): same K mapping, N = col
            int k0 = kk * 4 + hh * 2;
            v2f a2, b2;
            a2.x = Grow[k0];
            a2.y = Grow[k0 + 1];
            b2.x = P[k0 * PPITCH + wloc];
            b2.y = P[(k0 + 1) * PPITCH + wloc];
            acc = __builtin_amdgcn_wmma_f32_16x16x4_f32(
                /*neg_a=*/false, a2, /*neg_b=*/false, b2,
                /*c_mod=*/(short)0, acc, /*reuse_a=*/false, /*reuse_b=*/false);
        }
        // D layout: VGPR r -> freq row mt*16 + r + 8*hh, patch = wloc
        #pragma unroll
        for (int r = 0; r < 8; ++r) {
            int   kl = mt * 16 + hh * 8 + r;
            float lm = log10f(fabsf(acc[r]) + 1e-15f);
            #pragma unroll
            for (int m = 0; m < MBANDS; ++m)
                sum[m] = fmaf(F[m * KLPAD + kl], lm, sum[m]);
        }
    }

    // combine the two lane halves (complementary freq rows, same patch)
    #pragma unroll
    for (int m = 0; m < MBANDS; ++m)
        sum[m] += __shfl_xor(sum[m], 16, 32);

    if (hh == 0) {
        float* op = out + ((size_t)((n * HP + h) * HP) + wloc) * MBANDS;
        #pragma unroll
        for (int m = 0; m < MBANDS; ++m)
            op[m] = sum[m];
    }
}

extern "C" void kernel_launch(void* const* d_in, const int* in_sizes, int n_in,
                              void* d_out, int out_size, void* d_ws, size_t ws_size,
                              hipStream_t stream) {
    const float* x    = (const float*)d_in[0];   // [16,256,256,3] f32
    const float* filt = (const float*)d_in[1];   // [6,10,10] f32
    float*       out  = (float*)d_out;           // [16,128,128,6] f32

    const size_t smem_floats = 100 + (size_t)KLPAD * GPITCH
                             + (size_t)MBANDS * KLPAD + (size_t)100 * PPITCH;
    const size_t smem = smem_floats * sizeof(float);   // ~101 KB dynamic LDS

    dim3 grid(NIMG * HP);     // 2048 blocks: one patch row (n, h) each
    dim3 block(NTHREADS);     // 8 waves: one 16-patch WMMA N-tile per wave
    lfs_head_kernel<<<grid, block, smem, stream>>>(x, filt, out);
}